// SpectralConv2d_15006615734081
// MI455X (gfx1250) — compile-verified
//
#include <hip/hip_runtime.h>
#include <math.h>

typedef float v2f __attribute__((ext_vector_type(2)));
typedef float v8f __attribute__((ext_vector_type(8)));

// ---------------- workspace layout (floats) ----------------
constexpr size_t OFF_FWC  = 0;          // fwd W-axis twiddles cos   [ky][w]  32x256
constexpr size_t OFF_FWS  = 8192;       // fwd W-axis twiddles -sin  [ky][w]
constexpr size_t OFF_AHC  = 16384;      // fwd H-axis cos            [kx][h]  32x256
constexpr size_t OFF_AHS  = 24576;      // fwd H-axis +sin
constexpr size_t OFF_AHSN = 32768;      // fwd H-axis -sin
constexpr size_t OFF_EHC  = 40960;      // inv H-axis cos            [h][kx]  256x32
constexpr size_t OFF_EHS  = 49152;      // inv H-axis +sin
constexpr size_t OFF_EHSN = 57344;      // inv H-axis -sin
constexpr size_t OFF_EWC  = 65536;      // inv W-axis c*cos/65536    [w][ky]  256x32
constexpr size_t OFF_EWSN = 73728;      // inv W-axis -c*sin/65536
constexpr size_t OFF_WT   = 81920;                    // weights^T [mode][o][i]   4M floats
constexpr size_t OFF_S2R  = OFF_WT  + 4194304;        // Xft real  [mode][b*64+i] 1M
constexpr size_t OFF_S2I  = OFF_S2R + 1048576;        // Xft imag
constexpr size_t OFF_MR   = OFF_S2I + 1048576;        // mixed real [q][ky][kx]   1M
constexpr size_t OFF_MI   = OFF_MR  + 1048576;        // mixed imag

__device__ __forceinline__ v8f wmma_f32(v2f a, v2f b, v8f c) {
  // V_WMMA_F32_16X16X4_F32: D = A(16x4) * B(4x16) + C
  return __builtin_amdgcn_wmma_f32_16x16x4_f32(false, a, false, b, (short)0, c,
                                               false, false);
}

// ---------------- twiddle tables ----------------
__global__ __launch_bounds__(256) void k_init_tables(float* __restrict__ ws) {
  const int idx = blockIdx.x * 256 + threadIdx.x;       // 0..8191
  const float PI_128 = 0.0245436926061702596f;          // pi/128
  // interpretation A: [r][c], r in 0..31 (k-freq), c in 0..255 (space)
  int r = idx >> 8, c = idx & 255;
  float th = (float)((r * c) & 255) * PI_128;           // 2*pi*r*c/256 reduced
  float s = sinf(th), co = cosf(th);
  ws[OFF_FWC  + idx] = co;     // Aw real  (e^{-i th})
  ws[OFF_FWS  + idx] = -s;     // Aw imag
  ws[OFF_AHC  + idx] = co;     // Ah real
  ws[OFF_AHS  + idx] = s;      // -(Ah imag)
  ws[OFF_AHSN + idx] = -s;     // Ah imag
  // interpretation B: [row][col], row in 0..255 (space), col in 0..31 (k-freq)
  int row = idx >> 5, col = idx & 31;
  float th2 = (float)((row * col) & 255) * PI_128;
  float s2 = sinf(th2), c2 = cosf(th2);
  ws[OFF_EHC  + idx] = c2;     // Eh = e^{+i th}
  ws[OFF_EHS  + idx] = s2;
  ws[OFF_EHSN + idx] = -s2;
  float sc = (col == 0 ? 1.0f : 2.0f) * (1.0f / 65536.0f);  // Hermitian + irfft2 norm
  ws[OFF_EWC  + idx] = sc * c2;
  ws[OFF_EWSN + idx] = -sc * s2;
}

// weights [i][o][kx][ky] -> WT [mode=kx*32+ky][o][i]
__global__ __launch_bounds__(256) void k_transpose_w(const float* __restrict__ w,
                                                     float* __restrict__ wt) {
  size_t t = (size_t)blockIdx.x * 256 + threadIdx.x;    // = io*1024 + mode
  int mode = (int)(t & 1023);
  int io = (int)(t >> 10);
  int i = io >> 6, o = io & 63;
  wt[(size_t)mode * 4096 + (size_t)o * 64 + i] = w[t];
}

// ---------------- forward partial DFT: Xft = Ah * x * Aw ----------------
__global__ __launch_bounds__(256) void k_fwd_dft(const float* __restrict__ x,
                                                 const float* __restrict__ ws,
                                                 float* __restrict__ s2r,
                                                 float* __restrict__ s2i) {
  __shared__ float ldsXw[16384];   // Xw real [h*32+ky], imag at +8192
  const int tid = threadIdx.x;
  const int l = tid & 31, wv = tid >> 5;
  const int mrow = l & 15, kb = (l >> 4) * 2;
  const int p = blockIdx.x;                         // image index b*64+i

  const v8f z8 = {0.f, 0.f, 0.f, 0.f, 0.f, 0.f, 0.f, 0.f};

  // ---- GEMM1: Xw[256,32] = x[256,256] @ Aw[256,32] (A real, B complex) ----
  v8f accR[2][2], accI[2][2];
  #pragma unroll
  for (int a = 0; a < 2; ++a)
    #pragma unroll
    for (int b = 0; b < 2; ++b) { accR[a][b] = z8; accI[a][b] = z8; }

  const float* xr0 = x + (size_t)p * 65536 + (size_t)(wv * 32 + mrow) * 256;
  const float* xr1 = xr0 + 16 * 256;
  const float* bWc = ws + OFF_FWC + (size_t)mrow * 256;
  const float* bWs = ws + OFF_FWS + (size_t)mrow * 256;
  for (int k = 0; k < 256; k += 4) {
    v2f a0 = *(const v2f*)(xr0 + k + kb);
    v2f a1 = *(const v2f*)(xr1 + k + kb);
    #pragma unroll
    for (int nt = 0; nt < 2; ++nt) {
      v2f bc = *(const v2f*)(bWc + nt * 16 * 256 + k + kb);
      v2f bs = *(const v2f*)(bWs + nt * 16 * 256 + k + kb);
      accR[0][nt] = wmma_f32(a0, bc, accR[0][nt]);
      accI[0][nt] = wmma_f32(a0, bs, accI[0][nt]);
      accR[1][nt] = wmma_f32(a1, bc, accR[1][nt]);
      accI[1][nt] = wmma_f32(a1, bs, accI[1][nt]);
    }
  }
  #pragma unroll
  for (int mt = 0; mt < 2; ++mt)
    #pragma unroll
    for (int nt = 0; nt < 2; ++nt)
      #pragma unroll
      for (int v = 0; v < 8; ++v) {
        int h = wv * 32 + mt * 16 + 8 * (l >> 4) + v;
        int ky = nt * 16 + mrow;
        ldsXw[h * 32 + ky]        = accR[mt][nt][v];
        ldsXw[8192 + h * 32 + ky] = accI[mt][nt][v];
      }
  __syncthreads();

  // ---- GEMM2: Xft[32,32] = Ah[32,256] @ Xw (complex x complex), waves 0..3 ----
  if (wv < 4) {
    const int mt = wv >> 1, nt = wv & 1;
    v8f accXr = z8, accXi = z8;
    const float* aC  = ws + OFF_AHC  + (size_t)(mt * 16 + mrow) * 256;
    const float* aS  = ws + OFF_AHS  + (size_t)(mt * 16 + mrow) * 256;
    const float* aSN = ws + OFF_AHSN + (size_t)(mt * 16 + mrow) * 256;
    const int n = nt * 16 + mrow;
    for (int k = 0; k < 256; k += 4) {
      v2f ac  = *(const v2f*)(aC + k + kb);
      v2f as  = *(const v2f*)(aS + k + kb);
      v2f asn = *(const v2f*)(aSN + k + kb);
      v2f br, bi;
      br.x = ldsXw[(k + kb) * 32 + n];
      br.y = ldsXw[(k + kb + 1) * 32 + n];
      bi.x = ldsXw[8192 + (k + kb) * 32 + n];
      bi.y = ldsXw[8192 + (k + kb + 1) * 32 + n];
      // Xft_r = cos@Xwr + sin@Xwi ; Xft_i = cos@Xwi - sin@Xwr
      accXr = wmma_f32(ac, br, accXr);
      accXr = wmma_f32(as, bi, accXr);
      accXi = wmma_f32(ac, bi, accXi);
      accXi = wmma_f32(asn, br, accXi);
    }
    #pragma unroll
    for (int v = 0; v < 8; ++v) {
      int kx = mt * 16 + 8 * (l >> 4) + v;
      int ky = nt * 16 + mrow;
      size_t mode = (size_t)(kx * 32 + ky);
      s2r[mode * 1024 + p] = accXr[v];
      s2i[mode * 1024 + p] = accXi[v];
    }
  }
}

// ---------------- per-mode channel mixing: [16x64] @ [64x64] ----------------
__global__ __launch_bounds__(256) void k_mix(const float* __restrict__ s2r,
                                             const float* __restrict__ s2i,
                                             const float* __restrict__ wt,
                                             float* __restrict__ mr,
                                             float* __restrict__ mi) {
  const int tid = threadIdx.x;
  const int l = tid & 31, wv = tid >> 5;
  const int mrow = l & 15, kb = (l >> 4) * 2;
  const int mode = blockIdx.x * 8 + wv;
  const int kx = mode >> 5, ky = mode & 31;

  const v8f z8 = {0.f, 0.f, 0.f, 0.f, 0.f, 0.f, 0.f, 0.f};
  v8f accR[4], accI[4];
  #pragma unroll
  for (int nt = 0; nt < 4; ++nt) { accR[nt] = z8; accI[nt] = z8; }

  const float* Ar = s2r + (size_t)mode * 1024 + (size_t)mrow * 64;
  const float* Ai = s2i + (size_t)mode * 1024 + (size_t)mrow * 64;
  const float* Bw = wt + (size_t)mode * 4096 + (size_t)mrow * 64;
  for (int k = 0; k < 64; k += 4) {
    v2f ar = *(const v2f*)(Ar + k + kb);
    v2f ai = *(const v2f*)(Ai + k + kb);
    #pragma unroll
    for (int nt = 0; nt < 4; ++nt) {
      v2f b = *(const v2f*)(Bw + nt * 16 * 64 + k + kb);
      accR[nt] = wmma_f32(ar, b, accR[nt]);
      accI[nt] = wmma_f32(ai, b, accI[nt]);
    }
  }
  #pragma unroll
  for (int nt = 0; nt < 4; ++nt)
    #pragma unroll
    for (int v = 0; v < 8; ++v) {
      int b = 8 * (l >> 4) + v;
      int o = nt * 16 + mrow;
      size_t q = (size_t)(b * 64 + o);
      mr[q * 1024 + ky * 32 + kx] = accR[nt][v];
      mi[q * 1024 + ky * 32 + kx] = accI[nt][v];
    }
}

// ---------------- inverse partial DFT: out = Re(Eh * mixed * Ew) + bias ----------------
__global__ __launch_bounds__(256) void k_inv_dft(const float* __restrict__ ws,
                                                 const float* __restrict__ mr,
                                                 const float* __restrict__ mi,
                                                 const float* __restrict__ bias,
                                                 float* __restrict__ out) {
  __shared__ float ldsT[16384];  // T real [h*32+ky], imag at +8192
  const int tid = threadIdx.x;
  const int l = tid & 31, wv = tid >> 5;
  const int mrow = l & 15, kb = (l >> 4) * 2;
  const int q = blockIdx.x;                       // output image b*64+o

  const v8f z8 = {0.f, 0.f, 0.f, 0.f, 0.f, 0.f, 0.f, 0.f};

  // ---- Phase 1: T[256,32] = Eh[256,32] @ mixed[32,32] (complex x complex) ----
  v8f tR[2][2], tI[2][2];
  #pragma unroll
  for (int a = 0; a < 2; ++a)
    #pragma unroll
    for (int b = 0; b < 2; ++b) { tR[a][b] = z8; tI[a][b] = z8; }

  const float* Mr = mr + (size_t)q * 1024;   // [ky][kx]
  const float* Mi = mi + (size_t)q * 1024;
  for (int k = 0; k < 32; k += 4) {
    v2f ac[2], as_[2], asn[2];
    #pragma unroll
    for (int mm = 0; mm < 2; ++mm) {
      int h = (wv * 2 + mm) * 16 + mrow;
      ac[mm]  = *(const v2f*)(ws + OFF_EHC  + h * 32 + k + kb);
      as_[mm] = *(const v2f*)(ws + OFF_EHS  + h * 32 + k + kb);
      asn[mm] = *(const v2f*)(ws + OFF_EHSN + h * 32 + k + kb);
    }
    #pragma unroll
    for (int nt = 0; nt < 2; ++nt) {
      int n = nt * 16 + mrow;
      v2f br = *(const v2f*)(Mr + n * 32 + k + kb);
      v2f bi = *(const v2f*)(Mi + n * 32 + k + kb);
      #pragma unroll
      for (int mm = 0; mm < 2; ++mm) {
        // Tr = cos@Mr - sin@Mi ; Ti = cos@Mi + sin@Mr
        tR[mm][nt] = wmma_f32(ac[mm], br, tR[mm][nt]);
        tR[mm][nt] = wmma_f32(asn[mm], bi, tR[mm][nt]);
        tI[mm][nt] = wmma_f32(ac[mm], bi, tI[mm][nt]);
        tI[mm][nt] = wmma_f32(as_[mm], br, tI[mm][nt]);
      }
    }
  }
  #pragma unroll
  for (int mm = 0; mm < 2; ++mm)
    #pragma unroll
    for (int nt = 0; nt < 2; ++nt)
      #pragma unroll
      for (int v = 0; v < 8; ++v) {
        int h = (wv * 2 + mm) * 16 + 8 * (l >> 4) + v;
        int ky = nt * 16 + mrow;
        ldsT[h * 32 + ky]        = tR[mm][nt][v];
        ldsT[8192 + h * 32 + ky] = tI[mm][nt][v];
      }
  __syncthreads();

  // ---- Phase 2: out[256,256] = Tr @ Ewc + Ti @ EwsNeg, + bias ----
  const float bv = bias[q & 63];
  for (int nt = 0; nt < 16; ++nt) {
    v2f bc[8], bsn[8];
    #pragma unroll
    for (int kk = 0; kk < 8; ++kk) {
      int n = nt * 16 + mrow;
      bc[kk]  = *(const v2f*)(ws + OFF_EWC  + n * 32 + kk * 4 + kb);
      bsn[kk] = *(const v2f*)(ws + OFF_EWSN + n * 32 + kk * 4 + kb);
    }
    #pragma unroll
    for (int mm = 0; mm < 2; ++mm) {
      int mt = wv * 2 + mm;
      v8f acc = z8;
      #pragma unroll
      for (int kk = 0; kk < 8; ++kk) {
        int h = mt * 16 + mrow;
        v2f aTr = *(const v2f*)(&ldsT[h * 32 + kk * 4 + kb]);
        v2f aTi = *(const v2f*)(&ldsT[8192 + h * 32 + kk * 4 + kb]);
        acc = wmma_f32(aTr, bc[kk], acc);
        acc = wmma_f32(aTi, bsn[kk], acc);
      }
      #pragma unroll
      for (int v = 0; v < 8; ++v) {
        int h = mt * 16 + 8 * (l >> 4) + v;
        int wcol = nt * 16 + mrow;
        out[(size_t)q * 65536 + (size_t)h * 256 + wcol] = acc[v] + bv;
      }
    }
  }
}

extern "C" void kernel_launch(void* const* d_in, const int* in_sizes, int n_in,
                              void* d_out, int out_size, void* d_ws, size_t ws_size,
                              hipStream_t stream) {
  const float* x    = (const float*)d_in[0];  // [16,64,256,256]
  const float* w    = (const float*)d_in[1];  // [64,64,32,32]
  const float* bias = (const float*)d_in[2];  // [64]
  float* out = (float*)d_out;
  float* ws  = (float*)d_ws;

  k_init_tables<<<32, 256, 0, stream>>>(ws);
  k_transpose_w<<<16384, 256, 0, stream>>>(w, ws + OFF_WT);
  k_fwd_dft<<<1024, 256, 0, stream>>>(x, ws, ws + OFF_S2R, ws + OFF_S2I);
  k_mix<<<128, 256, 0, stream>>>(ws + OFF_S2R, ws + OFF_S2I, ws + OFF_WT,
                                 ws + OFF_MR, ws + OFF_MI);
  k_inv_dft<<<1024, 256, 0, stream>>>(ws, ws + OFF_MR, ws + OFF_MI, bias, out);
}